// RegressionModel_65481071401748
// MI455X (gfx1250) — compile-verified
//
#include <hip/hip_runtime.h>
#include <hip/hip_bf16.h>
#include <math.h>

#define N_WIRES 18
#define BATCH   64
#define KPAD    20          // K padded to 5 chunks of 4 for V_WMMA_F32_16X16X4_F32
#define NTHREADS 128        // 4 wave32 -> one 16-row M tile per wave

typedef __attribute__((ext_vector_type(2))) float v2f;
typedef __attribute__((ext_vector_type(8))) float v8f;

__global__ __launch_bounds__(NTHREADS)
void qnn_closed_form_kernel(const float* __restrict__ state,   // (64,18)
                            const float* __restrict__ params,  // (18,3)
                            const float* __restrict__ head_w,  // (1,18)
                            const float* __restrict__ head_b,  // (1,)
                            float* __restrict__ out)           // (64,)
{
    __shared__ float s_min[NTHREADS];
    __shared__ float s_max[NTHREADS];
    __shared__ float s_cos[BATCH][N_WIRES];
    __shared__ float s_sin[BATCH][N_WIRES];
    __shared__ float s_diag[N_WIRES];   // cos(p0)*cos(p1)
    __shared__ float s_off[N_WIRES];    // sin(p1)
    __shared__ float s_hw[KPAD];        // head_w zero-padded to K=20
    __shared__ float s_E[BATCH][KPAD];  // expvals, zero-padded K columns

    const int tid  = threadIdx.x;
    const int lane = tid & 31;
    const int wave = tid >> 5;

    // ---- 1. global min/max over all 64*18 encoder inputs ----
    float vmin =  1e30f, vmax = -1e30f;
    for (int i = tid; i < BATCH * N_WIRES; i += NTHREADS) {
        float v = state[i];
        vmin = fminf(vmin, v);
        vmax = fmaxf(vmax, v);
    }
    s_min[tid] = vmin;
    s_max[tid] = vmax;
    __syncthreads();
    for (int s = NTHREADS >> 1; s > 0; s >>= 1) {
        if (tid < s) {
            s_min[tid] = fminf(s_min[tid], s_min[tid + s]);
            s_max[tid] = fmaxf(s_max[tid], s_max[tid + s]);
        }
        __syncthreads();
    }
    const float mn = s_min[0];
    const float mx = s_max[0];
    const float scale = 6.28318530717958647692f / (mx - mn);

    // ---- 2. normalized angles -> cos/sin tables ----
    for (int i = tid; i < BATCH * N_WIRES; i += NTHREADS) {
        const int b = i / N_WIRES;
        const int w = i % N_WIRES;
        const float ang = (state[i] - mn) * scale - 3.14159265358979323846f;
        s_cos[b][w] = cosf(ang);
        s_sin[b][w] = sinf(ang);
    }

    // ---- 3. per-wire coefficients of U† Z U (RZ drops out) ----
    if (tid < N_WIRES) {
        const float p0 = params[tid * 3 + 0];
        const float p1 = params[tid * 3 + 1];
        s_diag[tid] = cosf(p0) * cosf(p1);
        s_off[tid]  = sinf(p1);
    }
    if (tid < KPAD)
        s_hw[tid] = (tid < N_WIRES) ? head_w[tid] : 0.0f;
    __syncthreads();

    // ---- 4. closed-form <Z_i> per (batch, wire) ----
    // <Z_i> = diag_i*cosT_i - off_i*sinT_i*cosT_{i-1}*cosT_{i+1}
    for (int i = tid; i < BATCH * N_WIRES; i += NTHREADS) {
        const int b = i / N_WIRES;
        const int w = i % N_WIRES;
        const float L = (w > 0)           ? s_cos[b][w - 1] : 1.0f;
        const float R = (w < N_WIRES - 1) ? s_cos[b][w + 1] : 1.0f;
        s_E[b][w] = s_diag[w] * s_cos[b][w] - s_off[w] * s_sin[b][w] * L * R;
    }
    // zero the padded K columns 18,19
    for (int i = tid; i < BATCH * (KPAD - N_WIRES); i += NTHREADS) {
        s_E[i / (KPAD - N_WIRES)][N_WIRES + i % (KPAD - N_WIRES)] = 0.0f;
    }
    __syncthreads();

    // ---- 5. head GEMV via V_WMMA_F32_16X16X4_F32 (full f32 precision) ----
    // M = 64 (4 tiles, one per wave), K = 20 (5 chunks of 4), N = 1 (col 0 of B)
    // A 16x4 f32 layout: lanes 0-15 -> M rows, VGPR0 = K {0 | 2}, VGPR1 = K {1 | 3}
    const int m  = (lane & 15) + wave * 16;  // global batch row
    const int hi = lane >> 4;                // 0: K pair {0,1}; 1: K pair {2,3}
    const int n  = lane & 15;                // B/D column

    v8f acc = {};
    #pragma unroll
    for (int k0 = 0; k0 < KPAD; k0 += 4) {
        v2f a, bmat;
        a.x = s_E[m][k0 + 2 * hi];
        a.y = s_E[m][k0 + 2 * hi + 1];
        bmat.x = (n == 0) ? s_hw[k0 + 2 * hi]     : 0.0f;
        bmat.y = (n == 0) ? s_hw[k0 + 2 * hi + 1] : 0.0f;
        // 8 args: (neg_a, A, neg_b, B, c_mod, C, reuse_a, reuse_b)
        acc = __builtin_amdgcn_wmma_f32_16x16x4_f32(
                  false, a, false, bmat, (short)0, acc, false, false);
    }

    // D layout: VGPR r -> M=r (lanes 0-15) / M=8+r (lanes 16-31), N = lane&15.
    // Only column N==0 is meaningful -> lanes 0 and 16 hold the outputs.
    const float bb = head_b[0];
    if (lane == 0) {
        #pragma unroll
        for (int r = 0; r < 8; ++r)
            out[wave * 16 + r] = acc[r] + bb;
    } else if (lane == 16) {
        #pragma unroll
        for (int r = 0; r < 8; ++r)
            out[wave * 16 + 8 + r] = acc[r] + bb;
    }
}

extern "C" void kernel_launch(void* const* d_in, const int* in_sizes, int n_in,
                              void* d_out, int out_size, void* d_ws, size_t ws_size,
                              hipStream_t stream) {
    const float* state  = (const float*)d_in[0];  // (64,18)
    const float* params = (const float*)d_in[1];  // (18,3)
    const float* head_w = (const float*)d_in[2];  // (1,18)
    const float* head_b = (const float*)d_in[3];  // (1,)
    float* out = (float*)d_out;                   // (64,)
    (void)in_sizes; (void)n_in; (void)out_size; (void)d_ws; (void)ws_size;

    qnn_closed_form_kernel<<<1, NTHREADS, 0, stream>>>(state, params, head_w, head_b, out);
}